// GeomPool_GNN_30837865185430
// MI455X (gfx1250) — compile-verified
//
#include <hip/hip_runtime.h>

#define HID 128
#define NB  64
#define DIN 1152

// d_ws layout (float offsets)
#define OFF_SUM 0            // [2][256][128] quadrant sums
#define OFF_CNT 65536        // [2][256] counts
#define OFF_REP 66048        // [64][1152]
#define OFF_H1  139776       // [64][256]
#define OFF_H2  156160       // [64][128]

typedef float v2f __attribute__((ext_vector_type(2)));
typedef float v8f __attribute__((ext_vector_type(8)));

// Guarantee the hardware f32 atomic (global_atomic_add_f32), not a CAS loop.
__device__ __forceinline__ void atomic_add_f32(float* p, float v) {
#if __has_builtin(__builtin_amdgcn_global_atomic_fadd_f32)
    __builtin_amdgcn_global_atomic_fadd_f32(p, v);
#else
    unsafeAtomicAdd(p, v);
#endif
}

__global__ __launch_bounds__(256) void k_zero(float* __restrict__ ws, int n) {
    int i = blockIdx.x * 256 + threadIdx.x;
    if (i < n) ws[i] = 0.0f;
}

// One wave per contiguous node chunk. Lane l owns columns [4l, 4l+3].
// Register-accumulate per quadrant; flush with f32 global atomics on batch change.
__global__ __launch_bounds__(256) void k_pool_accum(
    const float* __restrict__ x1, const float* __restrict__ x2,
    const float* __restrict__ pos1, const float* __restrict__ pos2,
    const int* __restrict__ batch1, const int* __restrict__ batch2,
    float* __restrict__ ws, int N)
{
    const int type = blockIdx.y;
    const float* __restrict__ x     = type ? x2 : x1;
    const float* __restrict__ pos   = type ? pos2 : pos1;
    const int*   __restrict__ batch = type ? batch2 : batch1;
    float* __restrict__ sum = ws + OFF_SUM + type * 32768;
    float* __restrict__ cnt = ws + OFF_CNT + type * 256;

    const int wid  = threadIdx.x >> 5;
    const int lane = threadIdx.x & 31;
    const int gw   = blockIdx.x * 8 + wid;
    const int nW   = gridDim.x * 8;
    const int chunk = (N + nW - 1) / nW;
    int n0 = gw * chunk;
    int n1 = n0 + chunk; if (n1 > N) n1 = N;
    if (n0 >= N) return;

    const int col = lane << 2;
    float4 acc[4];
    int qc[4];
#pragma unroll
    for (int q = 0; q < 4; ++q) { acc[q] = make_float4(0.f, 0.f, 0.f, 0.f); qc[q] = 0; }

    int cur_b = __builtin_amdgcn_readfirstlane(batch[n0]);

    auto flush = [&]() {
#pragma unroll
        for (int q = 0; q < 4; ++q) {
            if (qc[q]) {
                float* s = sum + (size_t)(cur_b * 4 + q) * HID + col;
                atomic_add_f32(s + 0, acc[q].x);
                atomic_add_f32(s + 1, acc[q].y);
                atomic_add_f32(s + 2, acc[q].z);
                atomic_add_f32(s + 3, acc[q].w);
                if (lane == 0) atomic_add_f32(cnt + cur_b * 4 + q, (float)qc[q]);
                acc[q] = make_float4(0.f, 0.f, 0.f, 0.f);
                qc[q] = 0;
            }
        }
    };

    for (int n = n0; n < n1; ++n) {
        int bb = __builtin_amdgcn_readfirstlane(batch[n]);
        if (bb != cur_b) { flush(); cur_b = bb; }
        float px = pos[2 * n + 0];
        float py = pos[2 * n + 1];
        int lbl = (py > px ? 1 : 0) + (px + py < 1200.0f ? 2 : 0);
        lbl = __builtin_amdgcn_readfirstlane(lbl);
        const float4 v = *reinterpret_cast<const float4*>(x + (size_t)n * HID + col);
#pragma unroll
        for (int q = 0; q < 4; ++q) {
            if (lbl == q) {
                acc[q].x += v.x; acc[q].y += v.y; acc[q].z += v.z; acc[q].w += v.w;
                qc[q]++;
            }
        }
    }
    flush();
}

// rep[b][c]: c<512 -> type1 quadrant means; 512..1023 -> type2; 1024.. -> x_global
__global__ __launch_bounds__(256) void k_finalize(
    const float* __restrict__ ws_sum, const float* __restrict__ ws_cnt,
    const float* __restrict__ xg, float* __restrict__ rep)
{
    int i = blockIdx.x * 256 + threadIdx.x;
    if (i >= NB * DIN) return;
    int b = i / DIN, c = i % DIN;
    float v;
    if (c < 1024) {
        int tt  = c >> 9;
        int cc  = c & 511;
        int seg = b * 4 + (cc >> 7);
        float s = ws_sum[tt * 32768 + seg * HID + (cc & 127)];
        float n = ws_cnt[tt * 256 + seg];
        v = s / fmaxf(n, 1.0f);
    } else {
        v = xg[b * HID + (c - 1024)];
    }
    rep[i] = v;
}

// C[64,N] = relu?(A[64,K] @ W[K,N] + bias). One 16x16 tile per wave via
// V_WMMA_F32_16X16X4_F32.  A lane layout: m=lane%16, k in {2*(lane/16), +1};
// B lane layout: n=lane%16, same k pair; D: row = r + 8*(lane/16), col = lane%16.
template <int K, int N, bool RELU>
__global__ __launch_bounds__(256) void k_gemm(
    const float* __restrict__ A, const float* __restrict__ W,
    const float* __restrict__ bias, float* __restrict__ C)
{
    const int wid  = threadIdx.x >> 5;
    const int lane = threadIdx.x & 31;
    const int t    = blockIdx.x * 8 + wid;      // tile id
    const int ntn  = N / 16;
    const int mt   = t / ntn, nt = t % ntn;
    const int mn   = lane & 15;                 // A row within tile / B col within tile
    const int kk   = (lane >> 4) << 1;          // 0 or 2

    const float* __restrict__ Arow = A + (size_t)(mt * 16 + mn) * K;
    const float* __restrict__ Bcol = W + nt * 16 + mn;

    v8f acc = {};
#pragma unroll 4
    for (int k0 = 0; k0 < K; k0 += 4) {
        v2f a, b;
        a.x = Arow[k0 + kk];
        a.y = Arow[k0 + kk + 1];
        b.x = Bcol[(size_t)(k0 + kk) * N];
        b.y = Bcol[(size_t)(k0 + kk + 1) * N];
        acc = __builtin_amdgcn_wmma_f32_16x16x4_f32(
            false, a, false, b, (short)0, acc, false, false);
    }

    const int row0 = mt * 16 + ((lane >> 4) << 3);
    const int ncol = nt * 16 + mn;
    const float bv = bias[ncol];
#pragma unroll
    for (int r = 0; r < 8; ++r) {
        float v = acc[r] + bv;
        if (RELU) v = v > 0.0f ? v : 0.0f;
        C[(size_t)(row0 + r) * N + ncol] = v;
    }
}

// out[64,2] = h2[64,128] @ W3[128,2] + b3   (N=2: scalar dot per thread)
__global__ __launch_bounds__(128) void k_out(
    const float* __restrict__ h2, const float* __restrict__ W3,
    const float* __restrict__ b3, float* __restrict__ out)
{
    int i = threadIdx.x;            // 0..127
    int b = i >> 1, o = i & 1;
    float s = b3[o];
#pragma unroll 4
    for (int k = 0; k < HID; ++k) s += h2[b * HID + k] * W3[k * 2 + o];
    out[i] = s;
}

extern "C" void kernel_launch(void* const* d_in, const int* in_sizes, int n_in,
                              void* d_out, int out_size, void* d_ws, size_t ws_size,
                              hipStream_t stream) {
    const float* x1     = (const float*)d_in[0];
    const float* x2     = (const float*)d_in[1];
    const float* xg     = (const float*)d_in[2];
    const float* pos1   = (const float*)d_in[3];
    const float* pos2   = (const float*)d_in[4];
    const int*   batch1 = (const int*)d_in[5];
    const int*   batch2 = (const int*)d_in[6];
    // d_in[7] = batch_global (identity; unused)
    const float* W1 = (const float*)d_in[8];
    const float* b1 = (const float*)d_in[9];
    const float* W2 = (const float*)d_in[10];
    const float* b2 = (const float*)d_in[11];
    const float* W3 = (const float*)d_in[12];
    const float* b3 = (const float*)d_in[13];
    float* ws  = (float*)d_ws;
    float* out = (float*)d_out;

    const int N = in_sizes[0] / HID;

    // zero accumulators (sums + counts)
    k_zero<<<(OFF_REP + 255) / 256, 256, 0, stream>>>(ws, OFF_REP);

    // streaming quadrant segment-sum, both node types (y dim)
    dim3 grid(1024, 2);
    k_pool_accum<<<grid, 256, 0, stream>>>(x1, x2, pos1, pos2, batch1, batch2, ws, N);

    // means + rep assembly
    k_finalize<<<(NB * DIN + 255) / 256, 256, 0, stream>>>(
        ws + OFF_SUM, ws + OFF_CNT, xg, ws + OFF_REP);

    // MLP
    k_gemm<DIN, 256, true><<<8, 256, 0, stream>>>(ws + OFF_REP, W1, b1, ws + OFF_H1);
    k_gemm<256, 128, true><<<4, 256, 0, stream>>>(ws + OFF_H1, W2, b2, ws + OFF_H2);
    k_out<<<1, 128, 0, stream>>>(ws + OFF_H2, W3, b3, out);
}